// SpatialDistributionLoss_30537217474922
// MI455X (gfx1250) — compile-verified
//
#include <hip/hip_runtime.h>
#include <hip/hip_bf16.h>

#define TH_F 0.5f
#define MAX_DIST_F 181.02f
#define HM 128
#define HW 16384           // 128*128
#define NPAIR 1088         // 64*17
#define BLK 256

#if defined(__has_builtin)
#if __has_builtin(__builtin_amdgcn_global_load_async_to_lds_b128)
#define USE_ASYNC_LDS 1
#endif
#if __has_builtin(__builtin_amdgcn_s_wait_asynccnt)
#define HAVE_WAIT_ASYNC 1
#endif
#endif

typedef int v4i_ __attribute__((ext_vector_type(4)));
typedef __attribute__((address_space(1))) v4i_* g_v4i_ptr;  // global
typedef __attribute__((address_space(3))) v4i_* l_v4i_ptr;  // LDS

__device__ __forceinline__ void wait_async_zero() {
#if defined(USE_ASYNC_LDS)
#if defined(HAVE_WAIT_ASYNC)
  __builtin_amdgcn_s_wait_asynccnt(0);
#else
  asm volatile("s_wait_asynccnt 0" ::: "memory");
#endif
#endif
}

// Stage 1: one block per (tensor, b, j) slice. Computes the per-slice
// normalized mean distance d and writes it to ws[bid].
__global__ __launch_bounds__(BLK) void sdl_stage1(const float* __restrict__ outp,
                                                  const float* __restrict__ targ,
                                                  float* __restrict__ ws) {
  __shared__ float tile[HW];      // 64 KB slice, loaded once from global
  __shared__ float swv[8];
  __shared__ int   swi[8];
  __shared__ float bmaxs;
  __shared__ int   bidxs;

  const int tid  = threadIdx.x;
  const int bid  = blockIdx.x;
  const int pair = (bid < NPAIR) ? bid : (bid - NPAIR);
  const float* __restrict__ src =
      ((bid < NPAIR) ? outp : targ) + (size_t)pair * HW;

  // ---- one-shot global -> LDS copy (async path on gfx1250, ASYNCcnt) ----
#if defined(USE_ASYNC_LDS)
  for (int k = 0; k < HW / (BLK * 4); ++k) {
    const int e = (k * BLK + tid) * 4;  // 16B-aligned element offset
    __builtin_amdgcn_global_load_async_to_lds_b128(
        (g_v4i_ptr)(src + e),
        (l_v4i_ptr)(&tile[e]),
        0, 0);
  }
  wait_async_zero();
#else
  for (int k = 0; k < HW / BLK; ++k) {
    const int p = k * BLK + tid;
    tile[p] = src[p];
  }
#endif
  __syncthreads();

  // ---- pass A: max + first-occurrence argmax (jnp.argmax semantics) ----
  float v  = -1.0f;
  int   vi = 0x7fffffff;
  for (int k = 0; k < HW / BLK; ++k) {
    const int p = k * BLK + tid;           // ascending per-thread -> '>' keeps first max
    const float x = tile[p];
    if (x > v) { v = x; vi = p; }
  }
  for (int off = 16; off > 0; off >>= 1) { // wave32 shuffle reduce
    const float ov = __shfl_down(v, off);
    const int   oi = __shfl_down(vi, off);
    if (ov > v || (ov == v && oi < vi)) { v = ov; vi = oi; }
  }
  const int wid = tid >> 5;
  if ((tid & 31) == 0) { swv[wid] = v; swi[wid] = vi; }
  __syncthreads();
  if (tid == 0) {
    float bv = swv[0]; int bi = swi[0];
    for (int w = 1; w < 8; ++w)
      if (swv[w] > bv || (swv[w] == bv && swi[w] < bi)) { bv = swv[w]; bi = swi[w]; }
    bmaxs = bv; bidxs = bi;
  }
  __syncthreads();

  const float maxv = bmaxs;
  const int   bi   = bidxs;
  const float ym   = (float)(bi >> 7);        // idx // 128
  const float xm   = (float)(bi & (HM - 1));  // idx %  128
  const float thr  = maxv * TH_F;

  // ---- pass B: masked distance sum + count, from LDS ----
  float s = 0.0f;
  int   c = 0;
  for (int k = 0; k < HW / BLK; ++k) {
    const int p = k * BLK + tid;
    const float x = tile[p];
    if (x > thr) {
      const float dy = (float)(p >> 7) - ym;
      const float dx = (float)(p & (HM - 1)) - xm;
      s += sqrtf(dy * dy + dx * dx);
      c += 1;
    }
  }
  for (int off = 16; off > 0; off >>= 1) {
    s += __shfl_down(s, off);
    c += __shfl_down(c, off);
  }
  if ((tid & 31) == 0) { swv[wid] = s; swi[wid] = c; }
  __syncthreads();
  if (tid == 0) {
    float ts = 0.0f; int tc = 0;
    for (int w = 0; w < 8; ++w) { ts += swv[w]; tc += swi[w]; }
    float d;
    if (tc > 0) d = (ts / (float)tc) / MAX_DIST_F;  // max(cnt,1)==cnt here
    else        d = 1.0f;
    if (!(maxv > 0.0f)) d = 0.0f;
    ws[bid] = d;
  }
}

// Stage 2: loss = sum |gt - pred| / J / B
__global__ __launch_bounds__(BLK) void sdl_stage2(const float* __restrict__ ws,
                                                  float* __restrict__ out) {
  __shared__ float sp[8];
  const int tid = threadIdx.x;
  float acc = 0.0f;
  for (int p = tid; p < NPAIR; p += BLK)
    acc += fabsf(ws[NPAIR + p] - ws[p]);    // |gt - pred|
  for (int off = 16; off > 0; off >>= 1) acc += __shfl_down(acc, off);
  if ((tid & 31) == 0) sp[tid >> 5] = acc;
  __syncthreads();
  if (tid == 0) {
    float t = 0.0f;
    for (int w = 0; w < 8; ++w) t += sp[w];
    out[0] = t / 17.0f / 64.0f;             // / J / B, reference order
  }
}

extern "C" void kernel_launch(void* const* d_in, const int* in_sizes, int n_in,
                              void* d_out, int out_size, void* d_ws, size_t ws_size,
                              hipStream_t stream) {
  (void)in_sizes; (void)n_in; (void)out_size; (void)ws_size;
  const float* outp = (const float*)d_in[0];
  const float* targ = (const float*)d_in[1];
  float* ws = (float*)d_ws;                 // needs 2*1088 floats = 8704 B

  sdl_stage1<<<2 * NPAIR, BLK, 0, stream>>>(outp, targ, ws);
  sdl_stage2<<<1, BLK, 0, stream>>>(ws, (float*)d_out);
}